// SparseLayer_42812234006677
// MI455X (gfx1250) — compile-verified
//
#include <hip/hip_runtime.h>

#define B_DIM 2048
#define D_DIM 512
#define H_DIM 1024
#define N_SAMPLE 100
#define K_TOP 51

typedef __attribute__((ext_vector_type(2))) float v2f;
typedef __attribute__((ext_vector_type(8))) float v8f;

// ---------------------------------------------------------------------------
// Kernel 1: eps_mean[b,d] = mean_s eps[s,b,d]   (dominant 419MB stream)
// ---------------------------------------------------------------------------
__global__ void __launch_bounds__(256)
eps_mean_kernel(const float* __restrict__ eps, float* __restrict__ em) {
    const size_t n = (size_t)B_DIM * D_DIM;            // 1,048,576
    const size_t i = ((size_t)blockIdx.x * blockDim.x + threadIdx.x) * 4;
    if (i >= n) return;
    float ax = 0.f, ay = 0.f, az = 0.f, aw = 0.f;
    #pragma unroll 4
    for (int s = 0; s < N_SAMPLE; ++s) {
        const float4 v = *reinterpret_cast<const float4*>(eps + (size_t)s * n + i);
        ax += v.x; ay += v.y; az += v.z; aw += v.w;
    }
    const float inv = 1.0f / (float)N_SAMPLE;
    float4 r; r.x = ax * inv; r.y = ay * inv; r.z = az * inv; r.w = aw * inv;
    *reinterpret_cast<float4*>(em + i) = r;
}

// ---------------------------------------------------------------------------
// Kernel 2: h = relu(batch @ W1^T + b1) via V_WMMA_F32_16X16X4_F32
//   one wave -> one 16x16 tile of h; 8 waves per block
// ---------------------------------------------------------------------------
__global__ void __launch_bounds__(256)
fc1_relu_wmma(const float* __restrict__ A,    // batch (B_DIM, D_DIM)
              const float* __restrict__ W1,   // (H_DIM, D_DIM)
              const float* __restrict__ b1,   // (H_DIM)
              float* __restrict__ Hout) {     // (B_DIM, H_DIM)
    const int K     = D_DIM;                  // 512
    const int wave  = threadIdx.x >> 5;
    const int lane  = threadIdx.x & 31;
    const int half  = lane >> 4;              // 0: K=0,1  1: K=2,3
    const int l     = lane & 15;

    const int tilesN = H_DIM / 16;            // 64
    const int gt = blockIdx.x * 8 + wave;
    const int tm = (gt / tilesN) * 16;
    const int tn = (gt % tilesN) * 16;

    // A[m,k] = batch[m*K + k];  Bmat[k,n] = W1[n*K + k]
    const float* Arow = A  + (size_t)(tm + l) * K + 2 * half;
    const float* Brow = W1 + (size_t)(tn + l) * K + 2 * half;

    v8f acc = {};
    #pragma unroll 8
    for (int k = 0; k < K; k += 4) {
        const v2f a = *reinterpret_cast<const v2f*>(Arow + k);
        const v2f b = *reinterpret_cast<const v2f*>(Brow + k);
        acc = __builtin_amdgcn_wmma_f32_16x16x4_f32(
            /*neg_a=*/false, a, /*neg_b=*/false, b,
            /*c_mod=*/(short)0, acc, /*reuse_a=*/false, /*reuse_b=*/false);
    }

    const int n = tn + l;
    const float bias = b1[n];
    #pragma unroll
    for (int r = 0; r < 8; ++r) {
        const int m = tm + half * 8 + r;
        float v = acc[r] + bias;
        Hout[(size_t)m * H_DIM + n] = v > 0.0f ? v : 0.0f;
    }
}

// ---------------------------------------------------------------------------
// Kernel 3: fused mu / log_var GEMMs + reparameterized mean + nonzero mask
//   c1 = h @ W21^T,  c2 = h @ W22^T  (shared A loads, two accumulators)
//   out = (mu + eps_mean * exp(0.5*log_var)) * (batch != 0)
// ---------------------------------------------------------------------------
__global__ void __launch_bounds__(256)
fc2_sample_wmma(const float* __restrict__ Hin,   // (B_DIM, H_DIM)
                const float* __restrict__ W21,   // (D_DIM, H_DIM)
                const float* __restrict__ b21,   // (D_DIM)
                const float* __restrict__ W22,   // (D_DIM, H_DIM)
                const float* __restrict__ b22,   // (D_DIM)
                const float* __restrict__ em,    // (B_DIM, D_DIM) eps mean
                const float* __restrict__ batch, // (B_DIM, D_DIM)
                float* __restrict__ Out) {       // (B_DIM, D_DIM)
    const int K    = H_DIM;                      // 1024
    const int wave = threadIdx.x >> 5;
    const int lane = threadIdx.x & 31;
    const int half = lane >> 4;
    const int l    = lane & 15;

    const int tilesN = D_DIM / 16;               // 32
    const int gt = blockIdx.x * 8 + wave;
    const int tm = (gt / tilesN) * 16;
    const int tn = (gt % tilesN) * 16;

    const float* Arow  = Hin + (size_t)(tm + l) * K + 2 * half;
    const float* B1row = W21 + (size_t)(tn + l) * K + 2 * half;
    const float* B2row = W22 + (size_t)(tn + l) * K + 2 * half;

    v8f c1 = {};
    v8f c2 = {};
    #pragma unroll 4
    for (int k = 0; k < K; k += 4) {
        const v2f a  = *reinterpret_cast<const v2f*>(Arow + k);
        const v2f b1v = *reinterpret_cast<const v2f*>(B1row + k);
        const v2f b2v = *reinterpret_cast<const v2f*>(B2row + k);
        c1 = __builtin_amdgcn_wmma_f32_16x16x4_f32(false, a, false, b1v,
                                                   (short)0, c1, false, false);
        c2 = __builtin_amdgcn_wmma_f32_16x16x4_f32(false, a, false, b2v,
                                                   (short)0, c2, false, false);
    }

    const int n = tn + l;
    const float bias1 = b21[n];
    const float bias2 = b22[n];
    #pragma unroll
    for (int r = 0; r < 8; ++r) {
        const int m = tm + half * 8 + r;
        const size_t idx = (size_t)m * D_DIM + n;
        const float mu = c1[r] + bias1;
        const float lv = c2[r] + bias2;
        const float sd = __expf(0.5f * lv);
        float op = mu + em[idx] * sd;
        op = (batch[idx] != 0.0f) ? op : 0.0f;
        Out[idx] = op;
    }
}

// ---------------------------------------------------------------------------
// Kernel 4: stable per-row top-K_TOP mask, in place on d_out.
//   keep element i iff #{j : row[j] > row[i] || (row[j]==row[i] && j<i)} < K
//   (exactly matches jax.lax.top_k index-order tie-breaking)
// ---------------------------------------------------------------------------
__global__ void __launch_bounds__(256)
topk_mask_kernel(float* __restrict__ out) {
    __shared__ float row[D_DIM];
    float* p = out + (size_t)blockIdx.x * D_DIM;
    for (int i = threadIdx.x; i < D_DIM; i += 256) row[i] = p[i];
    __syncthreads();
    for (int i = threadIdx.x; i < D_DIM; i += 256) {
        const float v = row[i];
        int beats = 0;
        #pragma unroll 8
        for (int j = 0; j < D_DIM; ++j) {
            const float u = row[j];
            beats += (u > v) | ((u == v) & (j < i));
        }
        p[i] = (beats < K_TOP) ? v : 0.0f;
    }
}

// ---------------------------------------------------------------------------
extern "C" void kernel_launch(void* const* d_in, const int* in_sizes, int n_in,
                              void* d_out, int out_size, void* d_ws, size_t ws_size,
                              hipStream_t stream) {
    const float* batch = (const float*)d_in[0];
    const float* W1    = (const float*)d_in[1];
    const float* b1    = (const float*)d_in[2];
    const float* W21   = (const float*)d_in[3];
    const float* b21   = (const float*)d_in[4];
    const float* W22   = (const float*)d_in[5];
    const float* b22   = (const float*)d_in[6];
    const float* eps   = (const float*)d_in[7];
    float* out = (float*)d_out;

    float* em = (float*)d_ws;                         // B*D floats (4 MB)
    float* h  = em + (size_t)B_DIM * D_DIM;           // B*H floats (8 MB)

    {   // eps mean: 419MB stream, float4 per thread
        const int threads = (B_DIM * D_DIM) / 4;      // 262144
        eps_mean_kernel<<<threads / 256, 256, 0, stream>>>(eps, em);
    }
    {   // GEMM1 + relu: 8192 tiles / 8 waves per block
        const int tiles = (B_DIM / 16) * (H_DIM / 16);
        fc1_relu_wmma<<<tiles / 8, 256, 0, stream>>>(batch, W1, b1, h);
    }
    {   // fused GEMM2 (mu & log_var) + sampling epilogue -> d_out
        const int tiles = (B_DIM / 16) * (D_DIM / 16);
        fc2_sample_wmma<<<tiles / 8, 256, 0, stream>>>(h, W21, b21, W22, b22,
                                                       em, batch, out);
    }
    // per-row top-k sparsification, in place
    topk_mask_kernel<<<B_DIM, 256, 0, stream>>>(out);
}